// Net_83262236000435
// MI455X (gfx1250) — compile-verified
//
#include <hip/hip_runtime.h>
#include <hip/hip_bf16.h>
#include <math.h>
#include <stdint.h>

typedef float v2f __attribute__((ext_vector_type(2)));
typedef float v8f __attribute__((ext_vector_type(8)));

#define NVERT 10000
#define TPB   256

// ---------------------------------------------------------------------------
// Generic fp32 WMMA GEMM: C = A@B (+ A2@B2) + bias.  A:[M,K] B:[K,N] row-major.
// One wave -> one 16x16 tile; 8 waves/block -> 128 output columns per block.
// Used for the small GEMMs (fc1, fc2, cheb dual-GEMMs).
// ---------------------------------------------------------------------------
__global__ void wmma_gemm_kernel(const float* __restrict__ A, const float* __restrict__ B,
                                 const float* __restrict__ A2, const float* __restrict__ B2,
                                 const float* __restrict__ bias, float* __restrict__ C,
                                 int M, int N, int K, int K2)
{
  const int wave  = threadIdx.x >> 5;
  const int lane  = threadIdx.x & 31;
  const int half  = lane >> 4;
  const int l16   = lane & 15;
  const int mbase = blockIdx.y << 4;
  const int nbase = (blockIdx.x << 7) + (wave << 4);
  if (mbase >= M || nbase >= N) return;   // wave-uniform exit (EXEC stays full)

  v8f acc = {};
  {
    const float* Ap  = A + (size_t)(mbase + l16) * K + (half << 1);
    const float* Bp0 = B + (size_t)(half << 1) * N + nbase + l16;
    for (int k = 0; k < K; k += 4) {
      v2f a = *(const v2f*)(Ap + k);
      const float* Bp = Bp0 + (size_t)k * N;
      v2f b; b.x = Bp[0]; b.y = Bp[N];
      acc = __builtin_amdgcn_wmma_f32_16x16x4_f32(false, a, false, b,
                                                  (short)0, acc, false, false);
    }
  }
  if (A2) {
    const float* Ap  = A2 + (size_t)(mbase + l16) * K2 + (half << 1);
    const float* Bp0 = B2 + (size_t)(half << 1) * N + nbase + l16;
    for (int k = 0; k < K2; k += 4) {
      v2f a = *(const v2f*)(Ap + k);
      const float* Bp = Bp0 + (size_t)k * N;
      v2f b; b.x = Bp[0]; b.y = Bp[N];
      acc = __builtin_amdgcn_wmma_f32_16x16x4_f32(false, a, false, b,
                                                  (short)0, acc, false, false);
    }
  }
  const float bv = bias ? bias[nbase + l16] : 0.0f;
#pragma unroll
  for (int r = 0; r < 8; ++r)
    C[(size_t)(mbase + r + (half << 3)) * N + nbase + l16] = acc[r] + bv;
}

// ---------------------------------------------------------------------------
// LDS-tiled fp32 WMMA GEMM for the big fc3 matmul (M=N=10000, K=256).
// Block tile 128x128, K-slice 32 (32 KB LDS), staged with CDNA5 async
// global->LDS copies (ASYNCcnt) and consumed via ds_load fragments.
// Per wave: 16 rows x 128 cols -> 8 v8f accumulators, 64 WMMA per K-slice.
// No early exits (EXEC must be all-ones for WMMA; barriers need all waves);
// OOB global addresses are clamped (valid duplicates), OOB results unstored.
// ---------------------------------------------------------------------------
#define BM 128
#define BN 128
#define BK 32

__global__ __launch_bounds__(TPB) void wmma_gemm_tiled_kernel(
    const float* __restrict__ A, const float* __restrict__ B,
    const float* __restrict__ bias, float* __restrict__ C,
    int M, int N, int K)
{
  __shared__ __align__(16) float As[BM * BK];   // 16 KB
  __shared__ __align__(16) float Bs[BK * BN];   // 16 KB

  const int tid  = threadIdx.x;
  const int wave = tid >> 5;
  const int lane = tid & 31;
  const int half = lane >> 4;
  const int l16  = lane & 15;
  const int mtile = blockIdx.y * BM;
  const int ntile = blockIdx.x * BN;

  v8f acc[8] = {};

  // LDS allocation-relative byte offsets (flat LDS aperture keeps the offset
  // in addr[31:0], which is what VDST of the async-to-LDS op expects).
  const unsigned asbase = (unsigned)(unsigned long long)(const void*)As;
  const unsigned bsbase = (unsigned)(unsigned long long)(const void*)Bs;

  for (int ks = 0; ks < K; ks += BK) {
    // ---- stage A tile [BM x BK]: 1024 float4, 4 async b128 issues/thread ----
#pragma unroll
    for (int i = 0; i < (BM * BK) / (4 * TPB); ++i) {
      int e   = tid + i * TPB;          // float4 index
      int row = e >> 3;                 // BK/4 = 8 float4 per row
      int c4  = e & 7;
      int grow = mtile + row; if (grow > M - 1) grow = M - 1;   // clamp: no OOB
      const float* gp = A + (size_t)grow * K + ks + (c4 << 2);
      unsigned loff = asbase + (unsigned)(((row * BK) + (c4 << 2)) << 2);
      asm volatile("global_load_async_to_lds_b128 %0, %1, off"
                   :: "v"(loff), "v"((unsigned long long)(uintptr_t)gp)
                   : "memory");
    }
    // ---- stage B tile [BK x BN] ----
#pragma unroll
    for (int i = 0; i < (BK * BN) / (4 * TPB); ++i) {
      int e   = tid + i * TPB;
      int row = e >> 5;                 // BN/4 = 32 float4 per row
      int c4  = e & 31;
      int gcol = ntile + (c4 << 2); if (gcol > N - 4) gcol = N - 4;  // clamp
      const float* gp = B + (size_t)(ks + row) * N + gcol;
      unsigned loff = bsbase + (unsigned)(((row * BN) + (c4 << 2)) << 2);
      asm volatile("global_load_async_to_lds_b128 %0, %1, off"
                   :: "v"(loff), "v"((unsigned long long)(uintptr_t)gp)
                   : "memory");
    }
    asm volatile("s_wait_asynccnt 0x0" ::: "memory");   // own wave's copies done
    __syncthreads();                                    // all waves' copies done

    // ---- compute: 8 k-steps x 8 n-tiles = 64 WMMA per wave per slice ----
    const int mrow = (wave << 4) + l16;
#pragma unroll
    for (int kk = 0; kk < BK; kk += 4) {
      v2f a = *(const v2f*)&As[mrow * BK + kk + (half << 1)];   // ds_load_b64
#pragma unroll
      for (int t = 0; t < 8; ++t) {
        v2f b;
        b.x = Bs[(kk + (half << 1)) * BN + (t << 4) + l16];
        b.y = Bs[(kk + (half << 1) + 1) * BN + (t << 4) + l16];
        acc[t] = __builtin_amdgcn_wmma_f32_16x16x4_f32(false, a, false, b,
                                                       (short)0, acc[t], false, false);
      }
    }
    __syncthreads();   // protect tiles before next slice overwrites them
  }

  // ---- guarded store ----
  const int mrow0 = mtile + (wave << 4);
#pragma unroll
  for (int t = 0; t < 8; ++t) {
    int col = ntile + (t << 4) + l16;
    if (col < N) {
      float bv = bias ? bias[col] : 0.f;
#pragma unroll
      for (int r = 0; r < 8; ++r) {
        int row = mrow0 + r + (half << 3);
        if (row < M) C[(size_t)row * N + col] = acc[t][r] + bv;
      }
    }
  }
}

// ---------------------------------------------------------------------------
// Split-K WMMA GEMM for 64x64 outputs with K=10000 (c = D@h, M = D@V).
// Deterministic: partials to separate buffers, summed by splitk_reduce.
// ---------------------------------------------------------------------------
__global__ void wmma_splitk_kernel(const float* __restrict__ A, const float* __restrict__ B,
                                   float* __restrict__ part, int K, int KC)
{
  const int lane  = threadIdx.x & 31;
  const int half  = lane >> 4;
  const int l16   = lane & 15;
  const int mbase = (blockIdx.x >> 2) << 4;
  const int nbase = (blockIdx.x & 3) << 4;
  const int k0    = blockIdx.y * KC;

  v8f acc = {};
  const float* Ap  = A + (size_t)(mbase + l16) * K + k0 + (half << 1);
  const float* Bp0 = B + (size_t)(k0 + (half << 1)) * 64 + nbase + l16;
  for (int k = 0; k < KC; k += 4) {
    v2f a = *(const v2f*)(Ap + k);
    const float* Bp = Bp0 + (size_t)k * 64;
    v2f b; b.x = Bp[0]; b.y = Bp[64];
    acc = __builtin_amdgcn_wmma_f32_16x16x4_f32(false, a, false, b,
                                                (short)0, acc, false, false);
  }
  float* Cp = part + (size_t)blockIdx.y * 4096;
#pragma unroll
  for (int r = 0; r < 8; ++r)
    Cp[(mbase + r + (half << 3)) * 64 + nbase + l16] = acc[r];
}

__global__ void splitk_reduce_kernel(const float* __restrict__ part, float* __restrict__ outm,
                                     int nchunk)
{
  int e = blockIdx.x * TPB + threadIdx.x;   // < 4096
  float s = 0.f;
  for (int i = 0; i < nchunk; ++i) s += part[(size_t)i * 4096 + e];
  outm[e] = s;
}

// ---------------------------------------------------------------------------
// Spectral micro-kernel (single block).  Given M=D@V [64,64], c=D@h [64,64],
// eigs[64], W[6,64,Cout] -> G[64,Cout], Wsum[64,Cout] such that
// cheb(h) = h@Wsum + V@G + cb.
// v0 = e.*c ; v1 = e.*(M v0) ; v_k = 2 e.*(M v_{k-1}) - v_{k-2}
// G = v0(W1-W3+W5) + 2v1(W2-W4) + 2v2(W3-W5) + 2v3 W4 + 2v4 W5
// Wsum = W0 - W2 + W4
// ---------------------------------------------------------------------------
__global__ void cheb_spectral_kernel(const float* __restrict__ Mm, const float* __restrict__ c,
                                     const float* __restrict__ eigs, const float* __restrict__ W,
                                     float* __restrict__ G, float* __restrict__ Wsum, int Cout)
{
  __shared__ float vb[3][64 * 64];          // 48 KB
  const int tid = threadIdx.x;
  const int GN  = 64 * Cout;
  const size_t wstride = (size_t)64 * Cout;

  for (int e = tid; e < 4096; e += TPB) vb[0][e] = eigs[e >> 6] * c[e];
  __syncthreads();

  float gacc[32];
  {
    int idx = 0;
    for (int e = tid; e < GN; e += TPB) gacc[idx++] = 0.f;
  }

  for (int k = 0; k < 5; ++k) {
    const int cur = k % 3;
    int idx = 0;
    for (int e = tid; e < GN; e += TPB, ++idx) {
      int r  = e / Cout;
      int co = e - r * Cout;
      float s = 0.f;
      for (int kk = 0; kk < 64; ++kk) {
        const float* Wk = W + (size_t)kk * Cout + co;
        float p;
        switch (k) {
          case 0:  p = Wk[1 * wstride] - Wk[3 * wstride] + Wk[5 * wstride]; break;
          case 1:  p = 2.f * (Wk[2 * wstride] - Wk[4 * wstride]);           break;
          case 2:  p = 2.f * (Wk[3 * wstride] - Wk[5 * wstride]);           break;
          case 3:  p = 2.f *  Wk[4 * wstride];                              break;
          default: p = 2.f *  Wk[5 * wstride];                              break;
        }
        s += vb[cur][(r << 6) + kk] * p;
      }
      gacc[idx] += s;
    }
    if (k < 4) {
      const int nxt = (k + 1) % 3, prev = (k + 2) % 3;
      for (int e = tid; e < 4096; e += TPB) {
        int r = e >> 6, cc = e & 63;
        float s = 0.f;
        for (int t = 0; t < 64; ++t) s += Mm[(r << 6) + t] * vb[cur][(t << 6) + cc];
        s *= eigs[r];
        vb[nxt][e] = (k == 0) ? s : 2.f * s - vb[prev][e];
      }
    }
    __syncthreads();
  }

  {
    int idx = 0;
    for (int e = tid; e < GN; e += TPB, ++idx) G[e] = gacc[idx];
  }
  for (int e = tid; e < GN; e += TPB)
    Wsum[e] = W[e] - W[2 * wstride + e] + W[4 * wstride + e];
}

// ---------------------------------------------------------------------------
// Per-column BN stats + apply, row-wise log-softmax.
// ---------------------------------------------------------------------------
__global__ void colstats_kernel(const float* __restrict__ X, float* __restrict__ stats,
                                int M, int C)
{
  __shared__ float s1[TPB], s2[TPB];
  const int j = blockIdx.x;
  float sum = 0.f, ss = 0.f;
  for (int i = threadIdx.x; i < M; i += TPB) {
    float v = X[(size_t)i * C + j];
    sum += v; ss += v * v;
  }
  s1[threadIdx.x] = sum; s2[threadIdx.x] = ss; __syncthreads();
  for (int o = TPB / 2; o > 0; o >>= 1) {
    if (threadIdx.x < o) { s1[threadIdx.x] += s1[threadIdx.x + o]; s2[threadIdx.x] += s2[threadIdx.x + o]; }
    __syncthreads();
  }
  if (threadIdx.x == 0) {
    float m = s1[0] / M;
    float var = s2[0] / M - m * m;
    stats[2 * j]     = m;
    stats[2 * j + 1] = rsqrtf(var + 1e-3f);
  }
}

__global__ void bnrelu_kernel(const float* __restrict__ X, const float* __restrict__ stats,
                              const float* __restrict__ g, const float* __restrict__ b,
                              float* __restrict__ Y, float* __restrict__ Y2, int M, int C)
{
  size_t i = (size_t)blockIdx.x * TPB + threadIdx.x;
  if (i >= (size_t)M * C) return;
  int j = (int)(i % C);
  float v = (X[i] - stats[2 * j]) * stats[2 * j + 1] * g[j] + b[j];
  v = fmaxf(v, 0.f);
  Y[i] = v;
  if (Y2) Y2[i] = v;
}

__global__ void rowlse_kernel(const float* __restrict__ L, float* __restrict__ rm,
                              float* __restrict__ rl, int N)
{
  __shared__ float sh[TPB];
  const float* p = L + (size_t)blockIdx.x * N;
  float m = -INFINITY;
  for (int j = threadIdx.x; j < N; j += TPB) m = fmaxf(m, p[j]);
  sh[threadIdx.x] = m; __syncthreads();
  for (int o = TPB / 2; o > 0; o >>= 1) {
    if (threadIdx.x < o) sh[threadIdx.x] = fmaxf(sh[threadIdx.x], sh[threadIdx.x + o]);
    __syncthreads();
  }
  m = sh[0]; __syncthreads();
  float s = 0.f;
  for (int j = threadIdx.x; j < N; j += TPB) s += expf(p[j] - m);
  sh[threadIdx.x] = s; __syncthreads();
  for (int o = TPB / 2; o > 0; o >>= 1) {
    if (threadIdx.x < o) sh[threadIdx.x] += sh[threadIdx.x + o];
    __syncthreads();
  }
  if (threadIdx.x == 0) { rm[blockIdx.x] = m; rl[blockIdx.x] = logf(sh[0]); }
}

__global__ void lsm_apply_kernel(float* __restrict__ L, const float* __restrict__ rm,
                                 const float* __restrict__ rl, int N)
{
  int row = blockIdx.y;
  int col = blockIdx.x * TPB + threadIdx.x;
  if (col < N) {
    size_t idx = (size_t)row * N + col;
    L[idx] = L[idx] - rm[row] - rl[row];
  }
}

// ---------------------------------------------------------------------------
extern "C" void kernel_launch(void* const* d_in, const int* in_sizes, int n_in,
                              void* d_out, int out_size, void* d_ws, size_t ws_size,
                              hipStream_t stream)
{
  (void)in_sizes; (void)n_in; (void)out_size; (void)ws_size;
  const float* x     = (const float*)d_in[0];
  const float* Vm    = (const float*)d_in[1];
  const float* Dm    = (const float*)d_in[2];
  const float* eigs  = (const float*)d_in[3];
  const float* fc1_w = (const float*)d_in[4];
  const float* fc1_b = (const float*)d_in[5];
  const float* bf1_g = (const float*)d_in[6];
  const float* bf1_b = (const float*)d_in[7];
  const float* W1    = (const float*)d_in[8];
  const float* cb1   = (const float*)d_in[9];
  const float* b1_g  = (const float*)d_in[10];
  const float* b1_b  = (const float*)d_in[11];
  const float* W2    = (const float*)d_in[12];
  const float* cb2   = (const float*)d_in[13];
  const float* b2_g  = (const float*)d_in[14];
  const float* b2_b  = (const float*)d_in[15];
  const float* W3    = (const float*)d_in[16];
  const float* cb3   = (const float*)d_in[17];
  const float* b3_g  = (const float*)d_in[18];
  const float* b3_b  = (const float*)d_in[19];
  const float* fc2_w = (const float*)d_in[20];
  const float* fc2_b = (const float*)d_in[21];
  const float* bf2_g = (const float*)d_in[22];
  const float* bf2_b = (const float*)d_in[23];
  const float* fc3_w = (const float*)d_in[24];
  const float* fc3_b = (const float*)d_in[25];

  float* out = (float*)d_out;
  float* ws  = (float*)d_ws;

  // workspace layout (floats)
  float* PRE  = ws;                  // 10000*256  pre-BN scratch
  float* HA   = PRE  + 2560000;      // 10000*128
  float* HB   = HA   + 1280000;      // 10000*128
  float* DESC = HB   + 1280000;      // 10000*256
  float* MM   = DESC + 2560000;      // 64*64   M = D@V
  float* CC   = MM   + 4096;         // 64*64   c = D@h
  float* GG   = CC   + 4096;         // 64*128
  float* WSU  = GG   + 8192;         // 64*128
  float* ST   = WSU  + 8192;         // 2*256 BN stats
  float* RM   = ST   + 512;          // 10000 row max
  float* RL   = RM   + 10000;        // 10000 row lse
  float* PART = RL   + 10000;        // 50*4096 split-K partials

  const int SPLITK = 50, KC = 200;   // 50 chunks * 200 = K=10000

  // ---- fc1 -> BN -> relu -> h1 (HA) ----
  wmma_gemm_kernel<<<dim3(1, 625), TPB, 0, stream>>>(x, fc1_w, nullptr, nullptr, fc1_b,
                                                     PRE, NVERT, 64, 352, 0);
  colstats_kernel<<<64, TPB, 0, stream>>>(PRE, ST, NVERT, 64);
  bnrelu_kernel<<<(NVERT * 64 + TPB - 1) / TPB, TPB, 0, stream>>>(PRE, ST, bf1_g, bf1_b,
                                                                  HA, nullptr, NVERT, 64);

  // ---- M = D @ V (shared by all three cheb layers) ----
  wmma_splitk_kernel<<<dim3(16, SPLITK), 32, 0, stream>>>(Dm, Vm, PART, NVERT, KC);
  splitk_reduce_kernel<<<16, TPB, 0, stream>>>(PART, MM, SPLITK);

  const float* Wl[3]  = {W1, W2, W3};
  const float* cbl[3] = {cb1, cb2, cb3};
  const float* gl[3]  = {b1_g, b2_g, b3_g};
  const float* bl[3]  = {b1_b, b2_b, b3_b};
  const int    col[3] = {64, 64, 128};

  float* hin = HA; float* hout = HB;
  for (int l = 0; l < 3; ++l) {
    const int Cout = col[l];
    wmma_splitk_kernel<<<dim3(16, SPLITK), 32, 0, stream>>>(Dm, hin, PART, NVERT, KC);
    splitk_reduce_kernel<<<16, TPB, 0, stream>>>(PART, CC, SPLITK);
    cheb_spectral_kernel<<<1, TPB, 0, stream>>>(MM, CC, eigs, Wl[l], GG, WSU, Cout);
    wmma_gemm_kernel<<<dim3((Cout + 127) / 128, 625), TPB, 0, stream>>>(hin, WSU, Vm, GG,
                                                                        cbl[l], PRE,
                                                                        NVERT, Cout, 64, 64);
    colstats_kernel<<<Cout, TPB, 0, stream>>>(PRE, ST, NVERT, Cout);
    bnrelu_kernel<<<(NVERT * Cout + TPB - 1) / TPB, TPB, 0, stream>>>(PRE, ST, gl[l], bl[l],
                                                                      hout, nullptr, NVERT, Cout);
    float* t = hin; hin = hout; hout = t;
  }
  // hin now holds h4 [10000,128]

  // ---- fc2 -> BN -> relu -> desc (ws copy + d_out tail copy) ----
  wmma_gemm_kernel<<<dim3(2, 625), TPB, 0, stream>>>(hin, fc2_w, nullptr, nullptr, fc2_b,
                                                     PRE, NVERT, 256, 128, 0);
  colstats_kernel<<<256, TPB, 0, stream>>>(PRE, ST, NVERT, 256);
  bnrelu_kernel<<<(NVERT * 256 + TPB - 1) / TPB, TPB, 0, stream>>>(
      PRE, ST, bf2_g, bf2_b, DESC, out + (size_t)NVERT * NVERT, NVERT, 256);

  // ---- fc3: LDS-tiled async-staged WMMA GEMM, logits straight into d_out ----
  wmma_gemm_tiled_kernel<<<dim3((NVERT + BN - 1) / BN, (NVERT + BM - 1) / BM), TPB, 0, stream>>>(
      DESC, fc3_w, fc3_b, out, NVERT, NVERT, 256);

  // ---- row-wise log_softmax in place ----
  rowlse_kernel<<<NVERT, TPB, 0, stream>>>(out, RM, RL, NVERT);
  lsm_apply_kernel<<<dim3((NVERT + TPB - 1) / TPB, NVERT), TPB, 0, stream>>>(out, RM, RL, NVERT);
}